// MoETransformer_66417374265886
// MI455X (gfx1250) — compile-verified
//
#include <hip/hip_runtime.h>
#include <math.h>

// Problem dims (fixed by the reference)
#define TTOK   4096      // B*S
#define DMODEL 1024
#define DFFN   4096
#define NEXP   8
#define NVOCAB 32000

// Tiling
#define MT   32          // token rows per FFN block
#define FK   64          // DFF chunk per fused iteration
#define OM   32          // token rows per out-proj block
#define XPAD 4           // LDS row padding (stride % 64 banks == 4 -> conflict-free)

// TDM-staged W2 chunking
#define W2CH       16                    // DFF rows per TDM chunk (FK/W2CH = 4 chunks)
#define W2LDSTRIDE 1032                  // effective padded row stride (floats)
#define W2BUF      (W2CH * DMODEL + 128) // 16384 data + 128 TDM pad dwords

typedef __attribute__((ext_vector_type(2))) float v2f;
typedef __attribute__((ext_vector_type(8))) float v8f;
typedef __attribute__((ext_vector_type(4))) unsigned int v4u;
typedef __attribute__((ext_vector_type(8))) int v8i;
typedef __attribute__((ext_vector_type(4))) int v4i;

__device__ __forceinline__ v8f wmma_f32x4(v2f a, v2f b, v8f c) {
  // D = A(16x4) * B(4x16) + C(16x16), all f32, wave32
  return __builtin_amdgcn_wmma_f32_16x16x4_f32(false, a, false, b, (short)0, c,
                                               false, false);
}

// Issue a TDM load of a [W2CH x DMODEL] f32 tile (row stride DMODEL) from
// `src` into LDS at byte address `lds_addr`, with LDS padding of 2 dwords
// after every 256 dwords (effective row stride W2LDSTRIDE floats).
__device__ __forceinline__ void tdm_load_w2(const float* src,
                                            unsigned lds_addr) {
  unsigned long long ga = (unsigned long long)(const void*)src;
  v4u g0;
  g0[0] = 1u;                                   // count=1, user mode
  g0[1] = lds_addr;                             // LDS byte address
  g0[2] = (unsigned)(ga & 0xFFFFFFFFull);       // global addr [31:0]
  g0[3] = (unsigned)((ga >> 32) & 0x01FFFFFFull) | (2u << 30);  // [56:32]|type=2
  v8i g1;
  g1[0] = (2 << 16)      // data_size = 4 bytes
        | (1 << 20)      // pad_enable
        | (7 << 22)      // pad_interval: 256 dwords
        | (1 << 25);     // pad_amount: 2 dwords
  g1[1] = (int)(((unsigned)DMODEL & 0xFFFFu) << 16);  // tensor_dim0[15:0]
  g1[2] = (int)(((unsigned)W2CH & 0xFFFFu) << 16);    // tensor_dim1[15:0]
  g1[3] = (int)(((unsigned)DMODEL & 0xFFFFu) << 16);  // tile_dim0
  g1[4] = W2CH;                                       // tile_dim1 (tile_dim2=0)
  g1[5] = DMODEL;                                     // tensor_dim0_stride[31:0]
  g1[6] = 0;
  g1[7] = 0;
  v4i z4 = {0, 0, 0, 0};
  v8i z8 = {0, 0, 0, 0, 0, 0, 0, 0};
  // amdgpu-toolchain (clang-23) 6-arg form: (g0, g1, g2, g3, extra, cpol)
  __builtin_amdgcn_tensor_load_to_lds(g0, g1, z4, z4, z8, 0);
}

// ---------------------------------------------------------------- embedding
__global__ void embed_gather(const int* __restrict__ tokens,
                             const float* __restrict__ emb,
                             float* __restrict__ x) {
  int t = blockIdx.x;
  int tok = tokens[t];
  const float4* src = (const float4*)(emb + (size_t)tok * DMODEL);
  float4* dst = (float4*)(x + (size_t)t * DMODEL);
  for (int i = threadIdx.x; i < DMODEL / 4; i += blockDim.x) dst[i] = src[i];
}

// ---------------------------------------------------------------- gating
__global__ __launch_bounds__(256) void gate_route(
    const float* __restrict__ x, const float* __restrict__ gw,
    const float* __restrict__ gb, int* __restrict__ counts,
    int* __restrict__ rtok, float* __restrict__ rgate,
    int* __restrict__ rpair) {
  int lane = threadIdx.x & 31;
  int wave = threadIdx.x >> 5;
  int t = blockIdx.x * 8 + wave;

  float acc[NEXP];
#pragma unroll
  for (int e = 0; e < NEXP; ++e) acc[e] = 0.f;

  const float* xr = x + (size_t)t * DMODEL;
  for (int d = lane; d < DMODEL; d += 32) {
    float xv = xr[d];
    const float4* g4 = (const float4*)(gw + (size_t)d * NEXP);
    float4 g0 = g4[0], g1 = g4[1];
    acc[0] += xv * g0.x; acc[1] += xv * g0.y;
    acc[2] += xv * g0.z; acc[3] += xv * g0.w;
    acc[4] += xv * g1.x; acc[5] += xv * g1.y;
    acc[6] += xv * g1.z; acc[7] += xv * g1.w;
  }
#pragma unroll
  for (int e = 0; e < NEXP; ++e) {
    float v = acc[e];
    for (int off = 16; off > 0; off >>= 1) v += __shfl_xor(v, off, 32);
    acc[e] = v + gb[e];
  }
  if (lane == 0) {
    int bi = 0; float bv = acc[0];
#pragma unroll
    for (int e = 1; e < NEXP; ++e)
      if (acc[e] > bv) { bv = acc[e]; bi = e; }
    int si = (bi == 0) ? 1 : 0; float sv = acc[si];
#pragma unroll
    for (int e = 0; e < NEXP; ++e)
      if (e != bi && acc[e] > sv) { sv = acc[e]; si = e; }
    float g1v = expf(sv - bv);            // softmax over {bv, sv}
    float inv = 1.f / (1.f + g1v);
    int s0 = atomicAdd(&counts[bi], 1);
    rtok[bi * TTOK + s0] = t; rgate[bi * TTOK + s0] = inv;
    rpair[bi * TTOK + s0] = t * 2;
    int s1 = atomicAdd(&counts[si], 1);
    rtok[si * TTOK + s1] = t; rgate[si * TTOK + s1] = g1v * inv;
    rpair[si * TTOK + s1] = t * 2 + 1;
  }
}

// ---------------------------------------------------------------- fused FFN
// Per (expert, 32-token tile): loop DFF in chunks of FK=64:
//   phase1: hs = relu(x_tile @ w1[:,chunk] + b1)          (hs stays in LDS)
//   phase2: acc += hs @ w2[chunk,:], with W2 streamed into LDS by the
//           Tensor Data Mover in 16-row chunks, double-buffered, synced
//           with s_wait_tensorcnt + workgroup barriers.
// Epilogue: y_slot[pair] = gate * (acc + b2)
__global__ __launch_bounds__(256) void moe_ffn(
    const float* __restrict__ x, const float* __restrict__ w1,
    const float* __restrict__ b1, const float* __restrict__ w2,
    const float* __restrict__ b2, const int* __restrict__ counts,
    const int* __restrict__ rtok, const float* __restrict__ rgate,
    const int* __restrict__ rpair, float* __restrict__ yslot) {
  __shared__ float xs[MT][DMODEL + XPAD];   // 32 x 1028 f32  (~131.5 KB)
  __shared__ float hs[MT][FK + XPAD];       // 32 x 68 f32    (~8.7 KB)
  __shared__ float w2s[2][W2BUF];           // 2 x 66 KB TDM ping-pong

  const int e = blockIdx.y;
  const int ne = counts[e];
  const int mbase = blockIdx.x * MT;
  if (mbase >= ne) return;
  const int tid = threadIdx.x;

  // stage x rows for this tile (zero-pad past ne)
  for (int idx = tid; idx < MT * (DMODEL / 4); idx += 256) {
    int r = idx >> 8;
    int c = (idx & 255) << 2;
    float4 v = make_float4(0.f, 0.f, 0.f, 0.f);
    int row = mbase + r;
    if (row < ne) {
      int tok = rtok[e * TTOK + row];
      v = *(const float4*)(x + (size_t)tok * DMODEL + c);
    }
    *(float4*)&xs[r][c] = v;
  }
  __syncthreads();

  const int wave = tid >> 5, lane = tid & 31;
  const int n16 = lane & 15, kh = lane >> 4;      // WMMA lane split
  const float* W1 = w1 + (size_t)e * DMODEL * DFFN;
  const float* W2 = w2 + (size_t)e * DFFN * DMODEL;
  const float* B1 = b1 + (size_t)e * DFFN;

  const unsigned w2lds0 = (unsigned)(uintptr_t)(void*)&w2s[0][0];
  const unsigned w2lds1 = (unsigned)(uintptr_t)(void*)&w2s[1][0];

  v8f acc[2][8];
  {
    v8f z = {};
#pragma unroll
    for (int i = 0; i < 2; ++i)
#pragma unroll
      for (int j = 0; j < 8; ++j) acc[i][j] = z;
  }
  const int msub = wave & 1;   // phase-1 16-row subtile
  const int nsub = wave >> 1;  // phase-1 16-col subtile of FK

  for (int f0 = 0; f0 < DFFN; f0 += FK) {
    // kick off the first two W2 chunks for this f-chunk (overlaps phase 1)
    if (wave == 0) {
      tdm_load_w2(W2 + (size_t)(f0 + 0 * W2CH) * DMODEL, w2lds0);
      tdm_load_w2(W2 + (size_t)(f0 + 1 * W2CH) * DMODEL, w2lds1);
    }

    // ---- phase 1: hs = relu(xs @ W1[:, f0:f0+FK] + b1) ----
    {
      const int hcol = f0 + nsub * 16;
      v8f h0 = {}, h1 = {};
      for (int k0 = 0; k0 < DMODEL; k0 += 8) {  // two accumulation chains
        v2f a, b;
        a.x = xs[msub * 16 + n16][k0 + 2 * kh];
        a.y = xs[msub * 16 + n16][k0 + 2 * kh + 1];
        b.x = W1[(size_t)(k0 + 2 * kh) * DFFN + hcol + n16];
        b.y = W1[(size_t)(k0 + 2 * kh + 1) * DFFN + hcol + n16];
        h0 = wmma_f32x4(a, b, h0);
        a.x = xs[msub * 16 + n16][k0 + 4 + 2 * kh];
        a.y = xs[msub * 16 + n16][k0 + 4 + 2 * kh + 1];
        b.x = W1[(size_t)(k0 + 4 + 2 * kh) * DFFN + hcol + n16];
        b.y = W1[(size_t)(k0 + 4 + 2 * kh + 1) * DFFN + hcol + n16];
        h1 = wmma_f32x4(a, b, h1);
      }
      v8f h = h0 + h1;
      float bias = B1[hcol + n16];
#pragma unroll
      for (int j = 0; j < 8; ++j) {
        float v = h[j] + bias;
        hs[msub * 16 + j + 8 * kh][nsub * 16 + n16] = fmaxf(v, 0.f);
      }
    }
    __syncthreads();   // hs ready

    // ---- phase 2: acc += hs(32xFK) @ W2[f0:f0+FK, wave's 128 cols] ----
    // TDM pad layout: stored col offset = col + 2*(col>>8)
    const int cadj = 2 * (wave >> 1);
#pragma unroll
    for (int j = 0; j < 4; ++j) {
      if (wave == 0) {
        if (j < 3) __builtin_amdgcn_s_wait_tensorcnt(1);  // chunk j landed
        else       __builtin_amdgcn_s_wait_tensorcnt(0);  // last chunk landed
      }
      __syncthreads();                       // publish chunk j to all waves
      const float* wb = &w2s[j & 1][0];
      for (int k0 = 0; k0 < W2CH; k0 += 4) {
        const int kk = j * W2CH + k0;        // row within FK for hs
        v2f a0, a1;
        a0.x = hs[n16][kk + 2 * kh];
        a0.y = hs[n16][kk + 2 * kh + 1];
        a1.x = hs[16 + n16][kk + 2 * kh];
        a1.y = hs[16 + n16][kk + 2 * kh + 1];
#pragma unroll
        for (int ns = 0; ns < 8; ++ns) {
          const int col = wave * 128 + ns * 16 + n16;
          v2f b;
          b.x = wb[(k0 + 2 * kh) * W2LDSTRIDE + col + cadj];
          b.y = wb[(k0 + 2 * kh + 1) * W2LDSTRIDE + col + cadj];
          acc[0][ns] = wmma_f32x4(a0, b, acc[0][ns]);
          acc[1][ns] = wmma_f32x4(a1, b, acc[1][ns]);
        }
      }
      __syncthreads();                       // chunk j fully consumed
      if (wave == 0 && j + 2 < 4)            // refill the freed buffer
        tdm_load_w2(W2 + (size_t)(f0 + (j + 2) * W2CH) * DMODEL,
                    (j & 1) ? w2lds1 : w2lds0);
    }
  }

  // epilogue: gate-scale + b2, write per-(token,k) slot
#pragma unroll
  for (int ms = 0; ms < 2; ++ms) {
#pragma unroll
    for (int j = 0; j < 8; ++j) {
      int rrow = mbase + ms * 16 + j + 8 * kh;
      if (rrow < ne) {
        int pair = rpair[e * TTOK + rrow];
        float g = rgate[e * TTOK + rrow];
#pragma unroll
        for (int ns = 0; ns < 8; ++ns) {
          int col = wave * 128 + ns * 16 + n16;
          yslot[(size_t)pair * DMODEL + col] =
              g * (acc[ms][ns][j] + b2[(size_t)e * DMODEL + col]);
        }
      }
    }
  }
}

// ---------------------------------------------------------------- combine
__global__ void combine_pairs(const float* __restrict__ ys,
                              float* __restrict__ xo) {
  size_t i = (size_t)blockIdx.x * blockDim.x + threadIdx.x;  // float4 index
  size_t t = i >> 8;              // 256 float4 per row
  size_t c = (i & 255) << 2;
  float4 va = *(const float4*)(ys + (2 * t) * DMODEL + c);
  float4 vb = *(const float4*)(ys + (2 * t + 1) * DMODEL + c);
  *(float4*)(xo + t * DMODEL + c) =
      make_float4(va.x + vb.x, va.y + vb.y, va.z + vb.z, va.w + vb.w);
}

// ---------------------------------------------------------------- out proj
__global__ __launch_bounds__(256) void out_proj(
    const float* __restrict__ x, const float* __restrict__ w,
    const float* __restrict__ bias, float* __restrict__ out) {
  __shared__ float xs[OM][DMODEL + XPAD];
  const int mbase = blockIdx.x * OM;
  const int nbase = blockIdx.y * 256;

  for (int idx = threadIdx.x; idx < OM * (DMODEL / 4); idx += 256) {
    int r = idx >> 8;
    int c = (idx & 255) << 2;
    *(float4*)&xs[r][c] = *(const float4*)(x + (size_t)(mbase + r) * DMODEL + c);
  }
  __syncthreads();

  const int wave = threadIdx.x >> 5, lane = threadIdx.x & 31;
  const int n16 = lane & 15, kh = lane >> 4;

  v8f acc[2][2];
  {
    v8f z = {};
    acc[0][0] = z; acc[0][1] = z; acc[1][0] = z; acc[1][1] = z;
  }
  for (int k0 = 0; k0 < DMODEL; k0 += 4) {
    v2f a0, a1;
    a0.x = xs[n16][k0 + 2 * kh];      a0.y = xs[n16][k0 + 2 * kh + 1];
    a1.x = xs[16 + n16][k0 + 2 * kh]; a1.y = xs[16 + n16][k0 + 2 * kh + 1];
#pragma unroll
    for (int ns = 0; ns < 2; ++ns) {
      int col = nbase + wave * 32 + ns * 16 + n16;
      v2f b;
      b.x = w[(size_t)(k0 + 2 * kh) * NVOCAB + col];
      b.y = w[(size_t)(k0 + 2 * kh + 1) * NVOCAB + col];
      acc[0][ns] = wmma_f32x4(a0, b, acc[0][ns]);
      acc[1][ns] = wmma_f32x4(a1, b, acc[1][ns]);
    }
  }
#pragma unroll
  for (int ms = 0; ms < 2; ++ms)
#pragma unroll
    for (int ns = 0; ns < 2; ++ns)
#pragma unroll
      for (int j = 0; j < 8; ++j) {
        int row = mbase + ms * 16 + j + 8 * kh;
        int col = nbase + wave * 32 + ns * 16 + n16;
        out[(size_t)row * NVOCAB + col] = acc[ms][ns][j] + bias[col];
      }
}

// ---------------------------------------------------------------- launcher
extern "C" void kernel_launch(void* const* d_in, const int* in_sizes, int n_in,
                              void* d_out, int out_size, void* d_ws,
                              size_t ws_size, hipStream_t stream) {
  (void)in_sizes; (void)n_in; (void)out_size; (void)ws_size;
  const int*   tokens = (const int*)d_in[0];
  const float* emb = (const float*)d_in[1];
  const float* g1w = (const float*)d_in[2];
  const float* g1b = (const float*)d_in[3];
  const float* w11 = (const float*)d_in[4];
  const float* b11 = (const float*)d_in[5];
  const float* w21 = (const float*)d_in[6];
  const float* b21 = (const float*)d_in[7];
  const float* g2w = (const float*)d_in[8];
  const float* g2b = (const float*)d_in[9];
  const float* w12 = (const float*)d_in[10];
  const float* b12 = (const float*)d_in[11];
  const float* w22 = (const float*)d_in[12];
  const float* b22 = (const float*)d_in[13];
  const float* ow  = (const float*)d_in[14];
  const float* ob  = (const float*)d_in[15];
  float* out = (float*)d_out;

  // workspace carve-up (~84 MB)
  char* p = (char*)d_ws;
  float* x0 = (float*)p; p += (size_t)TTOK * DMODEL * 4;
  float* x1 = (float*)p; p += (size_t)TTOK * DMODEL * 4;
  float* x2 = (float*)p; p += (size_t)TTOK * DMODEL * 4;
  float* ys = (float*)p; p += (size_t)2 * TTOK * DMODEL * 4;   // per (tok,k)
  int*   counts = (int*)p; p += 256;
  int*   rtok  = (int*)p;  p += (size_t)NEXP * TTOK * 4;
  float* rgate = (float*)p; p += (size_t)NEXP * TTOK * 4;
  int*   rpair = (int*)p;  p += (size_t)NEXP * TTOK * 4;

  embed_gather<<<TTOK, 256, 0, stream>>>(tokens, emb, x0);

  // ---- MoE layer 1 ----
  (void)hipMemsetAsync(counts, 0, NEXP * sizeof(int), stream);
  gate_route<<<TTOK / 8, 256, 0, stream>>>(x0, g1w, g1b, counts, rtok, rgate,
                                           rpair);
  moe_ffn<<<dim3(TTOK / MT, NEXP), 256, 0, stream>>>(
      x0, w11, b11, w21, b21, counts, rtok, rgate, rpair, ys);
  combine_pairs<<<TTOK, 256, 0, stream>>>(ys, x1);

  // ---- MoE layer 2 ----
  (void)hipMemsetAsync(counts, 0, NEXP * sizeof(int), stream);
  gate_route<<<TTOK / 8, 256, 0, stream>>>(x1, g2w, g2b, counts, rtok, rgate,
                                           rpair);
  moe_ffn<<<dim3(TTOK / MT, NEXP), 256, 0, stream>>>(
      x1, w12, b12, w22, b22, counts, rtok, rgate, rpair, ys);
  combine_pairs<<<TTOK, 256, 0, stream>>>(ys, x2);

  // ---- output projection ----
  out_proj<<<dim3(TTOK / OM, NVOCAB / 256), 256, 0, stream>>>(x2, ow, ob, out);
}